// MSEmbeddingNet_38319698215725
// MI455X (gfx1250) — compile-verified
//
#include <hip/hip_runtime.h>
#include <math.h>

typedef __attribute__((ext_vector_type(2))) float v2f;
typedef __attribute__((ext_vector_type(8))) float v8f;

#define N_SEQ 2000
#define NGATE 40   // 4*HID
#define HID   10
#define KDIM  32   // transform width

// ---------------------------------------------------------------------------
// Kernel 1: feedforward branches + gate-input GEMM (xg = T @ W_ih^T + b) via
// fp32 WMMA (V_WMMA_F32_16X16X4_F32). One block = 16 sequence rows, 3 waves,
// wave w computes output columns [16w, 16w+16) of the 48-wide padded tile.
// ---------------------------------------------------------------------------
__global__ __launch_bounds__(96)
void ff_wmma_kernel(const float* __restrict__ x,
                    const float* __restrict__ W_mz1, const float* __restrict__ b_mz1,
                    const float* __restrict__ W_mz2, const float* __restrict__ b_mz2,
                    const float* __restrict__ W_in1, const float* __restrict__ b_in1,
                    const float* __restrict__ W_in2, const float* __restrict__ b_in2,
                    const float* __restrict__ W_ih,
                    const float* __restrict__ b_ih, const float* __restrict__ b_hh,
                    float* __restrict__ xg)
{
    __shared__ float At[16][33];           // padded: avoid LDS bank conflicts
    const int tid = threadIdx.x;
    const int r0  = blockIdx.x * 16;       // 125 blocks * 16 rows == 2000 exactly

    // ---- compute transform tile [16 rows][32 cols] into LDS (plain VALU) ----
    for (int e = tid; e < 16 * KDIM; e += 96) {
        const int r  = e >> 5;
        const int cc = e & 31;
        const bool is_mz = (cc < 16);
        const float xv = x[(r0 + r) * 2 + (is_mz ? 0 : 1)];
        const float* W1 = is_mz ? W_mz1 : W_in1;
        const float* B1 = is_mz ? b_mz1 : b_in1;
        const float* W2 = is_mz ? W_mz2 : W_in2;
        const float* B2 = is_mz ? b_mz2 : b_in2;
        const int row2 = cc & 15;
        float acc = B2[row2];
        #pragma unroll
        for (int k = 0; k < 32; ++k) {
            float h1 = fmaf(xv, W1[k], B1[k]);
            h1 = h1 > 0.f ? h1 : 0.f;                 // relu (layer 1)
            acc = fmaf(W2[row2 * 32 + k], h1, acc);
        }
        At[r][cc] = acc > 0.f ? acc : 0.f;            // relu (layer 2)
    }
    __syncthreads();

    // ---- WMMA: C[16x16] += A[16x4] * B[4x16], 8 steps over K=32 ----
    const int lane = tid & 31;
    const int w    = tid >> 5;            // wave id -> N tile (uniform per wave)
    const int m    = lane & 15;           // A row / C column-lane index
    const int hi   = lane >> 4;           // K-half select per §7.12.2 layout
    const int n    = w * 16 + m;          // global gate column (may be >= 40)
    const bool valid_n = (n < NGATE);
    const int ncl  = valid_n ? n : 0;     // clamped addr -> branchless masking

    v8f c = {};
    #pragma unroll
    for (int k0 = 0; k0 < KDIM; k0 += 4) {
        const int kk = k0 + (hi << 1);
        v2f a, b;
        a.x = At[m][kk];
        a.y = At[m][kk + 1];
        float b0 = W_ih[ncl * KDIM + kk];
        float b1 = W_ih[ncl * KDIM + kk + 1];
        b.x = valid_n ? b0 : 0.f;
        b.y = valid_n ? b1 : 0.f;
        c = __builtin_amdgcn_wmma_f32_16x16x4_f32(
                false, a, false, b, (short)0, c, false, false);
    }

    if (valid_n) {
        const float bias = b_ih[n] + b_hh[n];
        #pragma unroll
        for (int r = 0; r < 8; ++r) {
            const int row = r + hi * 8;   // C layout: VGPR r -> M = r + 8*hi
            xg[(r0 + row) * NGATE + n] = c[r] + bias;
        }
    }
}

// ---------------------------------------------------------------------------
// Kernel 2: sequential LSTM scan, single wave32, zero barriers.
// Lane j owns gate j (and gate 32+(j&7) replicated); h broadcast via shuffles.
// ---------------------------------------------------------------------------
__device__ __forceinline__ float sigm_f(float v) {
    return 1.f / (1.f + __expf(-v));
}
__device__ __forceinline__ float tanh_f(float v) {
    const float ax = fabsf(v);
    const float e  = __expf(-2.f * ax);
    const float t  = (1.f - e) / (1.f + e);
    return copysignf(t, v);
}

__global__ __launch_bounds__(32)
void lstm_scan_kernel(const float* __restrict__ xg,
                      const float* __restrict__ W_hh,
                      float* __restrict__ out)
{
    const int lane = threadIdx.x;
    const int j0 = lane;             // gates 0..31
    const int j1 = 32 + (lane & 7);  // gates 32..39 (lanes 8..31 replicate)

    // Register-resident recurrent weights.
    float wa[HID], wb[HID];
    #pragma unroll
    for (int k = 0; k < HID; ++k) {
        wa[k] = W_hh[j0 * HID + k];
        wb[k] = W_hh[j1 * HID + k];
    }

    float hh[HID];
    #pragma unroll
    for (int k = 0; k < HID; ++k) hh[k] = 0.f;
    float cst = 0.f;                 // c[lane] for lanes 0..9

    // Software-pipelined gate-input loads (depth 1).
    float xga = xg[j0];
    float xgb = xg[j1];

    for (int t = 0; t < N_SEQ; ++t) {
        float xga_n = 0.f, xgb_n = 0.f;
        if (t + 1 < N_SEQ) {                       // uniform branch
            const float* p = xg + (size_t)(t + 1) * NGATE;
            xga_n = p[j0];
            xgb_n = p[j1];
        }
        if (t + 8 < N_SEQ)                         // global_prefetch_b8
            __builtin_prefetch(xg + (size_t)(t + 8) * NGATE, 0, 0);

        // g = xg_t + W_hh @ h  (10 FMAs per owned gate)
        float ga = xga, gb = xgb;
        #pragma unroll
        for (int k = 0; k < HID; ++k) {
            ga = fmaf(wa[k], hh[k], ga);
            gb = fmaf(wb[k], hh[k], gb);
        }

        // Gather the four gates for cell m = lane (valid for lanes 0..9).
        const float iv  = ga;                         // i[m] lives on lane m
        const float fv  = __shfl(ga, lane + 10, 32);  // f[m] on lane 10+m
        const float gv  = __shfl(ga, lane + 20, 32);  // g[m] on lane 20+m
        const float ova = __shfl(ga, lane + 30, 32);  // o[0..1] on lanes 30,31
        const float ovb = __shfl(gb, lane - 2, 32);   // o[2..9] on lanes 0..7
        const float ov  = (lane < 2) ? ova : ovb;

        const float cn = sigm_f(fv) * cst + sigm_f(iv) * tanh_f(gv);
        const float hn = sigm_f(ov) * tanh_f(cn);
        cst = cn;

        if (lane < HID) out[t * HID + lane] = hn;

        // Broadcast new h to every lane's register copy.
        #pragma unroll
        for (int k = 0; k < HID; ++k) hh[k] = __shfl(hn, k, 32);

        xga = xga_n;
        xgb = xgb_n;
    }
}

// ---------------------------------------------------------------------------
extern "C" void kernel_launch(void* const* d_in, const int* in_sizes, int n_in,
                              void* d_out, int out_size, void* d_ws, size_t ws_size,
                              hipStream_t stream)
{
    const float* x     = (const float*)d_in[0];
    const float* W_mz1 = (const float*)d_in[1];
    const float* b_mz1 = (const float*)d_in[2];
    const float* W_mz2 = (const float*)d_in[3];
    const float* b_mz2 = (const float*)d_in[4];
    const float* W_in1 = (const float*)d_in[5];
    const float* b_in1 = (const float*)d_in[6];
    const float* W_in2 = (const float*)d_in[7];
    const float* b_in2 = (const float*)d_in[8];
    const float* W_ih  = (const float*)d_in[9];
    const float* W_hh  = (const float*)d_in[10];
    const float* b_ih  = (const float*)d_in[11];
    const float* b_hh  = (const float*)d_in[12];

    float* xg = (float*)d_ws;          // [2000 x 40] = 320 KB scratch

    ff_wmma_kernel<<<N_SEQ / 16, 96, 0, stream>>>(
        x, W_mz1, b_mz1, W_mz2, b_mz2, W_in1, b_in1, W_in2, b_in2,
        W_ih, b_ih, b_hh, xg);

    lstm_scan_kernel<<<1, 32, 0, stream>>>(xg, W_hh, (float*)d_out);
}